// PointPillarsScatter_1726576853687
// MI455X (gfx1250) — compile-verified
//
#include <hip/hip_runtime.h>

// PointPillarsScatter for MI455X (gfx1250, wave32).
// Pass 1: zero 219MB canvas with 128-bit stores (RT temporal hint keeps most
//         of it L2-resident for the atomics in pass 2).
// Pass 2: one thread per (pillar, channel); coalesced 256B feature read per
//         pillar, wave-uniform scalar coord reads, no-return
//         global_atomic_add_f32 scatter (duplicates sum, matching .at[].add).

#define PPS_B   4
#define PPS_C   64
#define PPS_NY  496
#define PPS_NX  432

__global__ __launch_bounds__(256) void pps_scatter(const float* __restrict__ vf,
                                                   const int* __restrict__ coords,
                                                   float* __restrict__ out,
                                                   int P) {
    int tid = blockIdx.x * 256 + threadIdx.x;
    int p = tid >> 6;          // pillar index; wave-uniform (wave32: one wave
                               // covers 32 consecutive channels of one pillar)
    int c = tid & 63;          // channel index; lanes 0..31 -> consecutive c
    if (p >= P) return;

    // p is uniform across the wave (32-aligned groups of 64 tids map to one p).
    // readfirstlane makes that explicit so the 3 coord reads become scalar
    // (SMEM) loads instead of 3 redundant per-lane VMEM loads.
    int ps = __builtin_amdgcn_readfirstlane(p);
    int b = coords[ps * 4 + 0];
    int y = coords[ps * 4 + 2];
    int x = coords[ps * 4 + 3];

    float v = vf[(size_t)p * PPS_C + c];   // coalesced: 256B per pillar row

    const size_t plane = (size_t)PPS_NY * (size_t)PPS_NX;
    size_t idx = ((size_t)b * PPS_C + (size_t)c) * plane
               + (size_t)y * (size_t)PPS_NX + (size_t)x;

    // Hardware f32 atomic add, result discarded -> no-return
    // global_atomic_add_f32 (STOREcnt-tracked, RMW performed at L2).
    unsafeAtomicAdd(&out[idx], v);
}

__global__ __launch_bounds__(256) void pps_zero_f4(float4* __restrict__ out4,
                                                   unsigned long long n4) {
    unsigned long long i = (unsigned long long)blockIdx.x * 256ull + threadIdx.x;
    if (i < n4) {
        float4 z;
        z.x = 0.0f; z.y = 0.0f; z.z = 0.0f; z.w = 0.0f;
        out4[i] = z;                       // global_store_b128
    }
}

__global__ __launch_bounds__(32) void pps_zero_tail(float* __restrict__ out,
                                                    unsigned long long start,
                                                    unsigned long long n_total) {
    unsigned long long i = start + threadIdx.x;
    if (i < n_total) out[i] = 0.0f;
}

extern "C" void kernel_launch(void* const* d_in, const int* in_sizes, int n_in,
                              void* d_out, int out_size, void* d_ws, size_t ws_size,
                              hipStream_t stream) {
    const float* vf     = (const float*)d_in[0];   // (P, 64) f32
    const int*   coords = (const int*)d_in[1];     // (P, 4)  i32: b, z, y, x
    float*       out    = (float*)d_out;           // (B, C, NY, NX) f32

    int P = in_sizes[0] / PPS_C;

    unsigned long long n_total = (unsigned long long)out_size;
    unsigned long long n4 = n_total / 4ull;        // 13,713,408 for this shape
    unsigned int zero_blocks = (unsigned int)((n4 + 255ull) / 256ull);
    pps_zero_f4<<<zero_blocks, 256, 0, stream>>>((float4*)out, n4);

    unsigned long long rem = n_total - n4 * 4ull;  // 0 for this shape
    if (rem) {
        pps_zero_tail<<<1, 32, 0, stream>>>(out, n4 * 4ull, n_total);
    }

    long long total = (long long)P * PPS_C;
    unsigned int scat_blocks = (unsigned int)((total + 255) / 256);
    pps_scatter<<<scat_blocks, 256, 0, stream>>>(vf, coords, out, P);
}